// KVCacheHybrid_9242769622138
// MI455X (gfx1250) — compile-verified
//
#include <hip/hip_runtime.h>
#include <cstdint>

// ---------------------------------------------------------------------------
// KV-cache 4-bit quantize/update/requantize for MI455X (gfx1250).
// Bandwidth-bound (512 MB traffic -> ~22us floor @ 23.3 TB/s). Uses the CDNA5
// Tensor Data Mover to DMA each 64x128 f32 slice (strided 2D tile) into LDS,
// then register-resident two-pass affine quantization with wave32 reductions.
// Streaming (non-temporal) hints on both the TDM fill and the output stores:
// zero data reuse, working set >> 192 MB L2.
// ---------------------------------------------------------------------------

typedef unsigned int v4u __attribute__((ext_vector_type(4)));
typedef int          v4i __attribute__((ext_vector_type(4)));
typedef int          v8i __attribute__((ext_vector_type(8)));
typedef float        v4f __attribute__((ext_vector_type(4)));

namespace {
constexpr int Bc = 2, Hc = 32, Lc = 4096, Dc = 128, Sc = 16;
constexpr int ROWS  = Bc * Hc;          // 64 rows per slice (stride L*D or S*D)
constexpr int SLICE = ROWS * Dc;        // 8192 floats = 32 KB tile
constexpr int NTHREADS = 256;           // 8 wave32
constexpr int V4T = SLICE / 4 / NTHREADS;  // 8 float4 per thread
}

extern __shared__ __align__(16) float smem[];   // [0,SLICE): TDM tile @ LDS 0; then scratch

__device__ __forceinline__ void block_minmax(float lmn, float lmx, float* sred,
                                             float& omn, float& omx) {
  // wave32 butterfly reduction
  #pragma unroll
  for (int off = 16; off > 0; off >>= 1) {
    lmn = fminf(lmn, __shfl_xor(lmn, off, 32));
    lmx = fmaxf(lmx, __shfl_xor(lmx, off, 32));
  }
  const int wave = threadIdx.x >> 5;
  const int lane = threadIdx.x & 31;
  __syncthreads();                       // protect scratch reuse across rounds
  if (lane == 0) { sred[wave] = lmn; sred[8 + wave] = lmx; }
  __syncthreads();
  float mn = sred[0], mx = sred[8];
  #pragma unroll
  for (int w = 1; w < 8; ++w) { mn = fminf(mn, sred[w]); mx = fmaxf(mx, sred[8 + w]); }
  omn = mn; omx = mx;
}

__global__ __launch_bounds__(NTHREADS)
void kv4_update_kernel(const float* __restrict__ k_cache,
                       const float* __restrict__ v_cache,
                       const float* __restrict__ k_val,
                       const float* __restrict__ v_val,
                       const int*   __restrict__ input_pos,
                       float*       __restrict__ out) {
  const int l = blockIdx.x;              // slice index along L
  const int t = blockIdx.y;              // 0 = k, 1 = v
  const float* cache = (t == 0) ? k_cache : v_cache;
  const float* vnew  = (t == 0) ? k_val  : v_val;
  float* dst = out + (size_t)t * (size_t)Bc * Hc * Lc * Dc;

  // Scatter detection: whole slice replaced if l in input_pos (last match wins).
  int s_match = -1;
  for (int s = 0; s < Sc; ++s)
    if (input_pos[s] == l) s_match = s;
  const bool replaced = (s_match >= 0);

  const float* src = replaced ? (vnew + (size_t)s_match * Dc)
                              : (cache + (size_t)l * Dc);
  const unsigned long long row_stride =
      replaced ? (unsigned long long)(Sc * Dc) : (unsigned long long)(Lc * Dc);

  // ---- Tensor DMA descriptor: 2D tile, 64 rows x 128 f32, strided rows ----
  const unsigned long long ga = (unsigned long long)(uintptr_t)src;
  v4u g0 = {0u, 0u, 0u, 0u};
  g0[0] = 1u;                                            // count=1, user mode
  g0[1] = 0u;                                            // lds_addr = 0 (tile base)
  g0[2] = (unsigned)(ga & 0xFFFFFFFFull);                // global_addr[31:0]
  g0[3] = (unsigned)((ga >> 32) & 0x01FFFFFFull) | (2u << 30);  // addr[56:32] | type=2
  v8i g1 = {0, 0, 0, 0, 0, 0, 0, 0};
  g1[0] = 2 << 16;                                       // data_size = 4B; mask=0
  g1[1] = Dc << 16;                                      // tensor_dim0 = 128 (lo16)
  g1[2] = ROWS << 16;                                    // dim0 hi=0 | tensor_dim1 = 64 (lo16)
  g1[3] = Dc << 16;                                      // dim1 hi=0 | tile_dim0 = 128
  g1[4] = ROWS;                                          // tile_dim1 = 64, tile_dim2 = 0
  g1[5] = (int)(row_stride & 0xFFFFFFFFull);             // tensor_dim0_stride[31:0]
  g1[6] = (int)((row_stride >> 32) & 0xFFFFull);         // stride[47:32] | dim1_stride lo = 0
  g1[7] = 0;
  v4i g2 = {0, 0, 0, 0};
  v4i g3 = {0, 0, 0, 0};
  v8i g4 = {0, 0, 0, 0, 0, 0, 0, 0};                     // extra group (clang-23 6-arg form)

  if ((threadIdx.x >> 5) == 0) {                         // one wave issues the TDM op
    __builtin_amdgcn_tensor_load_to_lds(g0, g1, g2, g3, g4, /*cpol: TH_NT*/ 1);
    __builtin_amdgcn_s_wait_tensorcnt(0);
  }
  __syncthreads();

  const v4f* tile4 = (const v4f*)smem;
  float* sred = smem + SLICE;

  // Pull tile into registers; local min/max on the way.
  v4f v[V4T];
  float lmn = INFINITY, lmx = -INFINITY;
  #pragma unroll
  for (int k = 0; k < V4T; ++k) {
    v[k] = tile4[threadIdx.x + k * NTHREADS];
    lmn = fminf(lmn, fminf(fminf(v[k].x, v[k].y), fminf(v[k].z, v[k].w)));
    lmx = fmaxf(lmx, fmaxf(fmaxf(v[k].x, v[k].y), fmaxf(v[k].z, v[k].w)));
  }

  // Round A: only for non-replaced slices (quant->dequant of old cache).
  if (!replaced) {
    float mn1, mx1;
    block_minmax(lmn, lmx, sred, mn1, mx1);
    const float scale1 = fmaxf(mx1 - mn1, 1e-6f) / 15.0f;
    const float rs1 = 1.0f / scale1;
    const float z1 = mn1 + scale1 * 8.0f;
    lmn = INFINITY; lmx = -INFINITY;
    #pragma unroll
    for (int k = 0; k < V4T; ++k) {
      #pragma unroll
      for (int c = 0; c < 4; ++c) {
        float q  = rintf((v[k][c] - mn1) * rs1);         // round half-to-even
        q        = fminf(fmaxf(q, 0.0f), 15.0f);
        float x1 = (q - 8.0f) * scale1 + z1;             // dequant
        v[k][c] = x1;
        lmn = fminf(lmn, x1);
        lmx = fmaxf(lmx, x1);
      }
    }
  }

  // Round B: re-quantize updated slice, write reconstructed values.
  float mn2, mx2;
  block_minmax(lmn, lmx, sred, mn2, mx2);
  const float scale2 = fmaxf(mx2 - mn2, 1e-6f) / 15.0f;
  const float rs2 = 1.0f / scale2;
  const float z2 = mn2 + scale2 * 8.0f;

  #pragma unroll
  for (int k = 0; k < V4T; ++k) {
    const int i4  = threadIdx.x + k * NTHREADS;
    const int row = i4 >> 5;                             // 32 float4 per 128-f row
    const int col = (i4 & 31) << 2;
    v4f o;
    #pragma unroll
    for (int c = 0; c < 4; ++c) {
      float q = rintf((v[k][c] - mn2) * rs2);
      q       = fminf(fmaxf(q, 0.0f), 15.0f);
      o[c]    = (q - 8.0f) * scale2 + z2;
    }
    v4f* daddr = (v4f*)(dst + (size_t)row * (Lc * Dc) + (size_t)l * Dc + col);
    __builtin_nontemporal_store(o, daddr);               // streaming store (th:NT)
  }
}

extern "C" void kernel_launch(void* const* d_in, const int* in_sizes, int n_in,
                              void* d_out, int out_size, void* d_ws, size_t ws_size,
                              hipStream_t stream) {
  const float* k_cache = (const float*)d_in[0];
  const float* v_cache = (const float*)d_in[1];
  const float* k_val   = (const float*)d_in[2];
  const float* v_val   = (const float*)d_in[3];
  const int*   pos     = (const int*)d_in[4];
  float* out = (float*)d_out;

  dim3 grid(Lc, 2, 1);
  dim3 block(NTHREADS, 1, 1);
  size_t shmem = (size_t)SLICE * sizeof(float) + 64 * sizeof(float);
  kv4_update_kernel<<<grid, block, shmem, stream>>>(k_cache, v_cache, k_val, v_val, pos, out);
}